// BasicS2Conv_62337155334546
// MI455X (gfx1250) — compile-verified
//
#include <hip/hip_runtime.h>
#include <hip/hip_bf16.h>

// out[b,d,p,r] = sum_{c,k,a} W[d,c,kidx[k,r],aidx[a,r]] * x[b,c,k,p,a]
// Per r: GEMM  M=128(d) x N=4096(b,p) x K=19968(c,k,a), fp32 via
// V_WMMA_F32_16X16X4_F32. x staged into LDS by the Tensor Data Mover
// (tensor_load_to_lds, TENSORcnt), W gathered via LDS offset table.
// Each wave: 16(d) x 32(p) tile x 4 r values -> every B fragment feeds 4 WMMAs, every A fragment 2.

typedef __attribute__((ext_vector_type(2))) float v2f;
typedef __attribute__((ext_vector_type(8))) float v8f;
typedef __attribute__((ext_vector_type(4))) unsigned int u32x4;
typedef __attribute__((ext_vector_type(4))) int i32x4;
typedef __attribute__((ext_vector_type(8))) int i32x8;

#define DIM_B 4
#define DIM_C 128
#define DIM_D 128
#define DIM_K 13
#define DIM_A 12
#define DIM_P 1024
#define DIM_R 12
#define CC    2                        // c-chunk per LDS stage
#define CKA   (CC * DIM_K * DIM_A)     // 312 contraction steps per chunk
#define NT    32                       // p tile per wave (2 WMMA N-tiles)
#define RG    4                        // r values accumulated per pass

#define X_CSTR (DIM_K * DIM_P * DIM_A) // 159744
#define X_KSTR (DIM_P * DIM_A)         // 12288
#define X_BSTR (DIM_C * X_CSTR)        // 20447232
#define W_DSTR (DIM_C * 5 * DIM_A)     // 7680
#define ROW    (NT * DIM_A)            // 384 floats per (c',k) row
#define XS_ELEMS (CC * DIM_K * ROW)    // 9984 floats = 39.9KB
#define XS_VEC4  (XS_ELEMS / 4)        // 2496 float4s

__global__ __launch_bounds__(256) void s2conv_wmma_f32(
    const float* __restrict__ x, const float* __restrict__ W,
    const int* __restrict__ kidx3, const int* __restrict__ aidx3,
    float* __restrict__ out)
{
    __shared__ float xs[XS_ELEMS];     // [c'][k][p][a] natural layout
    __shared__ int   offt[RG * CKA];   // W gather offsets per r' and (c',k,a)

    const int tid  = threadIdx.x;
    const int lane = tid & 31;
    const int wave = tid >> 5;
    const int hi   = lane >> 4;        // lane half (0/1)
    const int nl   = lane & 15;        // N index within tile / M index
    const int b    = blockIdx.y;
    const int p0   = blockIdx.x * NT;
    const int r0   = blockIdx.z * RG;

    // offt[rp*CKA + (c'*13+k)*12+a] = c'*60 + kidx[k,r]*12 + aidx[a,r]
    for (int i = tid; i < RG * CKA; i += 256) {
        int rp  = i / CKA;
        int kk  = i - rp * CKA;
        int cp  = kk / (DIM_K * DIM_A);
        int rem = kk - cp * (DIM_K * DIM_A);
        int k   = rem / DIM_A;
        int a   = rem - k * DIM_A;
        int r   = r0 + rp;
        int ko  = kidx3[k * (DIM_A * DIM_A) + r]; // kidx3[k][0][r], 0..4
        int ao  = aidx3[a * DIM_A + r];           // aidx3[0][a][r], 0..11
        offt[i] = cp * (5 * DIM_A) + ko * DIM_A + ao;
    }

    v8f acc[RG][2];
    #pragma unroll
    for (int rp = 0; rp < RG; ++rp) {
        acc[rp][0] = (v8f){};
        acc[rp][1] = (v8f){};
    }

    const int d     = wave * 16 + nl;              // WMMA M -> output d row
    const int xbase = b * X_BSTR + p0 * DIM_A;     // multiple of 4 floats

#if __has_builtin(__builtin_amdgcn_tensor_load_to_lds)
    // TDM descriptor (ISA 8.3-8.6): 3D tile 384(f32) x 13(k) x 2(c'),
    // strides 12288 / 159744 elements; lands linearly in xs[].
    const unsigned lds_addr = (unsigned)(size_t)&xs[0];
    i32x8 g1;
    g1[0] = (int)(2u << 16);                            // data_size=4B
    g1[1] = (int)((unsigned)(X_KSTR & 0xFFFF) << 16);   // tensor_dim0 lo16
    g1[2] = (int)((unsigned)(X_KSTR >> 16) | (13u << 16));  // dim0 hi | dim1 lo
    g1[3] = (int)(384u << 16);                          // dim1 hi=0 | tile_dim0
    g1[4] = (int)(13u | (2u << 16));                    // tile_dim1 | tile_dim2
    g1[5] = (int)(unsigned)X_KSTR;                      // stride0 lo32
    g1[6] = (int)((unsigned)(X_CSTR & 0xFFFF) << 16);   // stride0 hi=0 | stride1 lo16
    g1[7] = (int)((unsigned)X_CSTR >> 16);              // stride1 bits 47:16
    i32x4 g2;
    g2[0] = DIM_C;                                      // tensor_dim2
    g2[1] = 1;                                          // tensor_dim3
    g2[2] = X_CSTR;                                     // tensor_dim2_stride lo32
    g2[3] = 0;                                          // stride2 hi | tile_dim3=0
    i32x4 g3 = {0, 0, 0, 0};
    i32x8 g4 = {0, 0, 0, 0, 0, 0, 0, 0};                // unused (ISA: no group 4)
#endif

    for (int c0 = 0; c0 < DIM_C; c0 += CC) {
        __syncthreads();  // protects offt (first iter) and xs reuse (later)

#if __has_builtin(__builtin_amdgcn_tensor_load_to_lds)
        if (wave == 0) {
            unsigned long long ga =
                (unsigned long long)(size_t)(x + xbase + c0 * X_CSTR);
            u32x4 g0;
            g0[0] = 1u;                                 // count=1, user desc
            g0[1] = lds_addr;                           // LDS byte address
            g0[2] = (unsigned)(ga & 0xFFFFFFFFu);       // global_addr lo
            g0[3] = (unsigned)((ga >> 32) & 0x01FFFFFFu) | (2u << 30); // hi|type=2
            __builtin_amdgcn_tensor_load_to_lds(g0, g1, g2, g3, g4, 0);
            __builtin_amdgcn_s_wait_tensorcnt(0);
        }
#else
        // Fallback: per-thread b128 staging.
        for (int i4 = tid; i4 < XS_VEC4; i4 += 256) {
            int ck = i4 / (ROW / 4);                   // c'*13 + k
            int t4 = i4 - ck * (ROW / 4);
            int cp = ck / DIM_K;
            int k  = ck - cp * DIM_K;
            int g  = xbase + (c0 + cp) * X_CSTR + k * X_KSTR + t4 * 4;
            float4 v = *(const float4*)(x + g);
            *(float4*)(xs + i4 * 4) = v;
            if (c0 + CC < DIM_C)
                __builtin_prefetch(x + g + CC * X_CSTR, 0, 0);
        }
#endif
        __syncthreads();

        const float* wb = W + d * W_DSTR + c0 * (5 * DIM_A);

        for (int ck = 0; ck < CC * DIM_K; ++ck) {
            const float* xr = xs + ck * ROW + nl * DIM_A;
            const int*   ot = offt + ck * DIM_A;
            #pragma unroll
            for (int a0 = 0; a0 < DIM_A; a0 += 4) {
                const int ab = a0 + 2 * hi;        // K rows {0,1}/{2,3} per half
                v2f bf0 = *(const v2f*)(xr + ab);            // ds_load_b64
                v2f bf1 = *(const v2f*)(xr + 192 + ab);      // second N tile
                #pragma unroll
                for (int rp = 0; rp < RG; ++rp) {
                    int2 o = *(const int2*)(ot + rp * CKA + ab); // ds_load_b64
                    v2f af;
                    af.x = wb[o.x];
                    af.y = wb[o.y];
                    acc[rp][0] = __builtin_amdgcn_wmma_f32_16x16x4_f32(
                        false, af, false, bf0, (short)0, acc[rp][0], false, false);
                    acc[rp][1] = __builtin_amdgcn_wmma_f32_16x16x4_f32(
                        false, af, false, bf1, (short)0, acc[rp][1], false, false);
                }
            }
        }
    }

    // C/D layout: VGPR j holds M=j (lanes 0-15) and M=j+8 (lanes 16-31), N=lane&15.
    const int dd = wave * 16 + hi * 8;
    #pragma unroll
    for (int n = 0; n < 2; ++n) {
        const int p = p0 + n * 16 + nl;
        #pragma unroll
        for (int j = 0; j < 8; ++j) {
            int o = ((b * DIM_D + dd + j) * DIM_P + p) * DIM_R + r0; // 16B aligned
            float4 v = make_float4(acc[0][n][j], acc[1][n][j],
                                   acc[2][n][j], acc[3][n][j]);
            *(float4*)(out + o) = v;
        }
    }
}

extern "C" void kernel_launch(void* const* d_in, const int* in_sizes, int n_in,
                              void* d_out, int out_size, void* d_ws, size_t ws_size,
                              hipStream_t stream) {
    const float* x     = (const float*)d_in[0];
    const float* W     = (const float*)d_in[1];
    const int*   kidx3 = (const int*)d_in[2];
    const int*   aidx3 = (const int*)d_in[3];
    float*       out   = (float*)d_out;
    (void)in_sizes; (void)n_in; (void)out_size; (void)d_ws; (void)ws_size;

    dim3 grid(DIM_P / NT, DIM_B, DIM_R / RG); // (32, 4, 3)
    s2conv_wmma_f32<<<grid, 256, 0, stream>>>(x, W, kidx3, aidx3, out);
}